// FastEWRLSLevel_27281632264623
// MI455X (gfx1250) — compile-verified
//
#include <hip/hip_runtime.h>
#include <cstdint>

// Problem geometry (compile-time constants from the reference)
#define B_      32
#define T_      4096
#define C_      512
#define LCH     64                 // timesteps per chunk
#define NC      (T_ / LCH)         // 64 chunks over T
#define TROWS   16                 // timesteps per TDM tile (tile = 16 x 512 f32 = 32 KB)
#define NTILE   (LCH / TROWS)      // 4 tiles per chunk
#define SEGS    4                  // T-segments for L2 residency (64 MB of x each)
#define CPS     (NC / SEGS)        // 16 chunks per segment
#define THREADS 128                // 4 waves; each thread owns 4 channels (float4)

typedef unsigned int u32;
typedef __attribute__((ext_vector_type(4))) unsigned int v4u;
typedef __attribute__((ext_vector_type(8))) int v8i;
typedef __attribute__((ext_vector_type(4))) int v4i;

// ---- Tensor Data Mover: load a (TROWS x C_) f32 tile (row stride C_) into LDS ----
// D# built per CDNA5 ISA 8.3/8.4:
//   group0: count=1 | lds_addr | global_addr[56:0] | type=2
//   group1: data_size=4B | tensor_dim0=C_ | tensor_dim1=T_ | tile_dim0=C_
//           tile_dim1=TROWS | tensor_dim0_stride=C_
// This toolchain's builtin takes the 6-arg form:
//   (v4u group0, v8i group1, v4i group2, v4i group3, v8i pad, i32 cpol)
__device__ __forceinline__ void tdm_load_tile(u32 lds_off, const float* gsrc) {
  uint64_t ga = (uint64_t)(uintptr_t)gsrc;
  v4u g0;
  g0.x = 1u;                                              // count=1, user mode
  g0.y = lds_off;                                         // lds_addr (bytes)
  g0.z = (u32)ga;                                         // global_addr[31:0]
  g0.w = (u32)((ga >> 32) & 0x01ffffffu) | (2u << 30);    // global_addr[56:32] | type=2
  v8i g1;
  g1.s0 = (int)(2u << 16);            // data_size = 2 (4 bytes); no multicast/pad/iterate
  g1.s1 = (int)((u32)C_ << 16);       // tensor_dim0[15:0] @ bits[63:48]
  g1.s2 = (int)((u32)T_ << 16);       // tensor_dim0[31:16]=0; tensor_dim1[15:0] @ [95:80]
  g1.s3 = (int)((u32)C_ << 16);       // tensor_dim1[31:16]=0; tile_dim0 = C_
  g1.s4 = TROWS;                      // tile_dim1 = TROWS; tile_dim2 = 0 (2D)
  g1.s5 = C_;                         // tensor_dim0_stride[31:0]
  g1.s6 = 0;                          // stride hi / dim1_stride (unused, 2D)
  g1.s7 = 0;
  v4i z4 = 0;                         // groups 2/3 unused (2D tensor)
  v8i z8 = 0;
  __builtin_amdgcn_tensor_load_to_lds(g0, g1, z4, z4, z8, 0);
}

__device__ __forceinline__ float lam_of(const float* logit, int c) {
  float s = 1.0f / (1.0f + expf(-logit[c]));
  return fminf(fmaxf(s, 1.0e-4f), 1.0f - 1.0e-4f);
}

// ---------------- Phase 1: per-chunk weighted partial sums m_j ----------------
// m_j[b,c] = sum_{k=0..LCH-1} lam^(LCH-1-k) * x[b, t0+k, c]
__global__ __launch_bounds__(THREADS) void ewrls_phase1(
    const float* __restrict__ x, const float* __restrict__ logit,
    float* __restrict__ m_seg, int seg) {
  __shared__ __align__(16) float buf[2][TROWS * C_];
  const int b  = blockIdx.x / CPS;
  const int lj = blockIdx.x % CPS;
  const int j  = seg * CPS + lj;
  const int t0 = j * LCH;
  const int c0 = (int)threadIdx.x * 4;
  const float* src = x + ((size_t)b * T_ + t0) * C_;
  const bool wave0 = (__builtin_amdgcn_readfirstlane((int)(threadIdx.x >> 5)) == 0);

  if (wave0) {
    tdm_load_tile((u32)(uintptr_t)&buf[0][0], src);
    tdm_load_tile((u32)(uintptr_t)&buf[1][0], src + (size_t)TROWS * C_);
  }

  const float lam0 = lam_of(logit, c0 + 0), lam1 = lam_of(logit, c0 + 1);
  const float lam2 = lam_of(logit, c0 + 2), lam3 = lam_of(logit, c0 + 3);
  float m0 = 0.f, m1 = 0.f, m2 = 0.f, m3 = 0.f;

  for (int tile = 0; tile < NTILE; ++tile) {
    if (wave0) {
      if (tile < NTILE - 1) __builtin_amdgcn_s_wait_tensorcnt(1);  // tile done, next in flight
      else                  __builtin_amdgcn_s_wait_tensorcnt(0);
    }
    __syncthreads();
    const float* lbuf = &buf[tile & 1][0];
#pragma unroll
    for (int r = 0; r < TROWS; ++r) {
      float4 xv = *(const float4*)(lbuf + r * C_ + c0);
      m0 = fmaf(lam0, m0, xv.x);
      m1 = fmaf(lam1, m1, xv.y);
      m2 = fmaf(lam2, m2, xv.z);
      m3 = fmaf(lam3, m3, xv.w);
    }
    __syncthreads();  // everyone done reading buf[tile&1] before DMA overwrites it
    if (wave0 && (tile + 2) < NTILE)
      tdm_load_tile((u32)(uintptr_t)&buf[tile & 1][0],
                    src + (size_t)(tile + 2) * TROWS * C_);
  }
  float4 mv = make_float4(m0, m1, m2, m3);
  *(float4*)(m_seg + ((size_t)(b * CPS + lj)) * C_ + c0) = mv;
}

// ---------------- Phase 2: tiny sequential scan over chunk carries ----------------
// carry_in[j] = N_{t0-1};  carry' = lam^LCH * carry + m_j.  Running carry lives in rc[].
__global__ __launch_bounds__(256) void ewrls_phase2(
    const float* __restrict__ logit, const float* __restrict__ m_seg,
    float* __restrict__ carry_seg, float* __restrict__ rc, int seg) {
  const int idx = blockIdx.x * 256 + (int)threadIdx.x;   // 0 .. B_*C_-1
  const int b = idx / C_;
  const int c = idx - b * C_;
  const float lam = lam_of(logit, c);
  const float A = powf(lam, (float)LCH);
  float carry = (seg == 0) ? 0.0f : rc[idx];
#pragma unroll
  for (int lj = 0; lj < CPS; ++lj) {
    const size_t o = ((size_t)(b * CPS + lj)) * C_ + c;
    carry_seg[o] = carry;
    carry = fmaf(A, carry, m_seg[o]);
  }
  rc[idx] = carry;
}

// ---------------- Phase 3: reproduce outputs theta_t = N_t * (1-lam)/(1-lam^{t+1}) ----------------
__global__ __launch_bounds__(THREADS) void ewrls_phase3(
    const float* __restrict__ x, const float* __restrict__ logit,
    const float* __restrict__ carry_seg, float* __restrict__ out, int seg) {
  __shared__ __align__(16) float buf[2][TROWS * C_];
  const int b  = blockIdx.x / CPS;
  const int lj = blockIdx.x % CPS;
  const int j  = seg * CPS + lj;
  const int t0 = j * LCH;
  const int c0 = (int)threadIdx.x * 4;
  const float* src = x + ((size_t)b * T_ + t0) * C_;
  const bool wave0 = (__builtin_amdgcn_readfirstlane((int)(threadIdx.x >> 5)) == 0);

  if (wave0) {
    tdm_load_tile((u32)(uintptr_t)&buf[0][0], src);
    tdm_load_tile((u32)(uintptr_t)&buf[1][0], src + (size_t)TROWS * C_);
  }

  float lam[4], n[4], lamp[4], oml[4];
#pragma unroll
  for (int i = 0; i < 4; ++i) {
    lam[i]  = lam_of(logit, c0 + i);
    oml[i]  = 1.0f - lam[i];
    lamp[i] = powf(lam[i], (float)(t0 + 1));   // lam^{t+1} at chunk entry
    n[i]    = carry_seg[((size_t)(b * CPS + lj)) * C_ + c0 + i];
  }

  for (int tile = 0; tile < NTILE; ++tile) {
    if (wave0) {
      if (tile < NTILE - 1) __builtin_amdgcn_s_wait_tensorcnt(1);
      else                  __builtin_amdgcn_s_wait_tensorcnt(0);
    }
    __syncthreads();
    const float* lbuf = &buf[tile & 1][0];
    float* orow = out + ((size_t)b * T_ + (size_t)(t0 + tile * TROWS)) * C_ + c0;
#pragma unroll
    for (int r = 0; r < TROWS; ++r) {
      float4 xv = *(const float4*)(lbuf + r * C_ + c0);
      float4 ov;
      n[0] = fmaf(lam[0], n[0], xv.x);
      ov.x = n[0] * (oml[0] * __builtin_amdgcn_rcpf(1.0f - lamp[0]));
      lamp[0] *= lam[0];
      n[1] = fmaf(lam[1], n[1], xv.y);
      ov.y = n[1] * (oml[1] * __builtin_amdgcn_rcpf(1.0f - lamp[1]));
      lamp[1] *= lam[1];
      n[2] = fmaf(lam[2], n[2], xv.z);
      ov.z = n[2] * (oml[2] * __builtin_amdgcn_rcpf(1.0f - lamp[2]));
      lamp[2] *= lam[2];
      n[3] = fmaf(lam[3], n[3], xv.w);
      ov.w = n[3] * (oml[3] * __builtin_amdgcn_rcpf(1.0f - lamp[3]));
      lamp[3] *= lam[3];
      *(float4*)orow = ov;
      orow += C_;
    }
    __syncthreads();
    if (wave0 && (tile + 2) < NTILE)
      tdm_load_tile((u32)(uintptr_t)&buf[tile & 1][0],
                    src + (size_t)(tile + 2) * TROWS * C_);
  }
}

extern "C" void kernel_launch(void* const* d_in, const int* in_sizes, int n_in,
                              void* d_out, int out_size, void* d_ws, size_t ws_size,
                              hipStream_t stream) {
  (void)in_sizes; (void)n_in; (void)out_size; (void)ws_size;
  const float* x     = (const float*)d_in[0];
  const float* logit = (const float*)d_in[1];
  float* out = (float*)d_out;

  // Workspace layout (floats): m_seg[B*CPS*C] | carry_seg[B*CPS*C] | rc[B*C]  (~2.1 MB)
  float* m_seg     = (float*)d_ws;
  float* carry_seg = m_seg + (size_t)B_ * CPS * C_;
  float* rc        = carry_seg + (size_t)B_ * CPS * C_;

  // Segment T so phase3's re-read of x hits the 192MB L2 primed by phase1 (64MB/segment).
  for (int s = 0; s < SEGS; ++s) {
    hipLaunchKernelGGL(ewrls_phase1, dim3(B_ * CPS), dim3(THREADS), 0, stream,
                       x, logit, m_seg, s);
    hipLaunchKernelGGL(ewrls_phase2, dim3((B_ * C_) / 256), dim3(256), 0, stream,
                       logit, m_seg, carry_seg, rc, s);
    hipLaunchKernelGGL(ewrls_phase3, dim3(B_ * CPS), dim3(THREADS), 0, stream,
                       x, logit, carry_seg, out, s);
  }
}